// GeodesicSolver_70377334112307
// MI455X (gfx1250) — compile-verified
//
#include <hip/hip_runtime.h>

// GeodesicSolver: 30 fused momentum-SGD sweeps of an FD path-length gradient.
// One workgroup (1024 threads = 32 wave32) per batch element; path tile lives
// in LDS (33 KB of the 320 KB WGP LDS), velocity lives in registers.
// ||a +- eps*e_d|| = sqrt(||a||^2 +- 2*eps*a_d + eps^2), and b[t] = a[t+1],
// so each sweep is: (1) per-segment squared norms via wave32 xor-reduction,
// (2) per-(t,d) gradient + momentum update, (3) write-back. 3 barriers/iter.

#define T_PTS   256
#define DIM     32
#define ITERS   30
#define FD_EPS  1e-4f

#if defined(__gfx1250__)
#  if __has_builtin(__builtin_amdgcn_global_load_async_to_lds_b128)
#    define HAVE_ASYNC_LD 1
#  endif
#  if __has_builtin(__builtin_amdgcn_global_store_async_from_lds_b128)
#    define HAVE_ASYNC_ST 1
#  endif
#  if __has_builtin(__builtin_amdgcn_s_wait_asynccnt)
#    define WAIT_ASYNC() __builtin_amdgcn_s_wait_asynccnt(0)
#  else
#    define WAIT_ASYNC() asm volatile("s_wait_asynccnt 0" ::: "memory")
#  endif
#endif

// Builtin parameter type per hipcc diagnostic:
//   '__attribute__((__vector_size__(4 * sizeof(int)))) int __device__ *'
// i.e. v4i in the global (AS1) address space; LDS side in AS3.
typedef int v4i __attribute__((vector_size(4 * sizeof(int))));
typedef __attribute__((address_space(1))) v4i g_v4i;   // global int4
typedef __attribute__((address_space(3))) v4i l_v4i;   // LDS int4

__global__ __launch_bounds__(1024, 1)
void GeodesicSolver_kernel(const float* __restrict__ path,
                           const float* __restrict__ step_p,
                           const float* __restrict__ mom_p,
                           float* __restrict__ out)
{
    __shared__ float p_lds[T_PTS * DIM];   // 32 KB: current path for this batch
    __shared__ float ns2[T_PTS];           // 1 KB: ns2[s] = ||p[s]-p[s-1]||^2, s=1..255

    const int tid  = threadIdx.x;
    const int lane = tid & 31;             // dimension d
    const int wv   = tid >> 5;             // wave id 0..31
    const size_t base = (size_t)blockIdx.x * (T_PTS * DIM);

    const float step = step_p[0];
    const float mom  = mom_p[0];

    // ---- stage path tile global -> LDS (8 floats / thread) ----
    {
        const float* g = path + base + (size_t)tid * 8;
        float*       l = &p_lds[tid * 8];
#if defined(HAVE_ASYNC_LD)
        // CDNA5 async DMA into LDS; tracked by ASYNCcnt.
        __builtin_amdgcn_global_load_async_to_lds_b128((g_v4i*)g, (l_v4i*)l, 0,  0);
        __builtin_amdgcn_global_load_async_to_lds_b128((g_v4i*)g, (l_v4i*)l, 16, 0);
        WAIT_ASYNC();
#else
        const float4* g4 = (const float4*)g;
        float4*       l4 = (float4*)l;
        l4[0] = g4[0];
        l4[1] = g4[1];
#endif
    }
    __syncthreads();

    float vloc[8];                         // velocity for owned rows t = 1+wv+32k
    float pnew[8];
#pragma unroll
    for (int k = 0; k < 8; ++k) vloc[k] = 0.0f;

    const float e2    = FD_EPS * FD_EPS;
    const float te    = 2.0f * FD_EPS;
    const float inv2e = 1.0f / (2.0f * FD_EPS);

    for (int it = 0; it < ITERS; ++it) {
        // ---- phase 1: segment squared norms (wave handles segments s = 1+wv+32k) ----
#pragma unroll
        for (int k = 0; k < 8; ++k) {
            const int s = 1 + wv + 32 * k;
            if (s <= T_PTS - 1) {
                const float seg = p_lds[s * DIM + lane] - p_lds[(s - 1) * DIM + lane];
                float sq = seg * seg;
                sq += __shfl_xor(sq, 16, 32);
                sq += __shfl_xor(sq,  8, 32);
                sq += __shfl_xor(sq,  4, 32);
                sq += __shfl_xor(sq,  2, 32);
                sq += __shfl_xor(sq,  1, 32);
                if (lane == 0) ns2[s] = sq;
            }
        }
        __syncthreads();

        // ---- phase 2: gradient + momentum update for owned interior rows ----
#pragma unroll
        for (int k = 0; k < 8; ++k) {
            const int t = 1 + wv + 32 * k;
            if (t <= T_PTS - 2) {
                const int idx = t * DIM + lane;
                const float pc  = p_lds[idx];
                const float ad  = pc - p_lds[idx - DIM];        // a[t] component d
                const float bd  = p_lds[idx + DIM] - pc;        // b[t] component d
                const float na2 = ns2[t];
                const float nb2 = ns2[t + 1];
                const float g =
                    ((__builtin_sqrtf(na2 + te * ad + e2) - __builtin_sqrtf(na2 - te * ad + e2)) +
                     (__builtin_sqrtf(nb2 - te * bd + e2) - __builtin_sqrtf(nb2 + te * bd + e2))) * inv2e;
                const float vn = mom * vloc[k] - step * g;
                vloc[k] = vn;
                pnew[k] = pc + vn;
            }
        }
        __syncthreads();

        // ---- phase 3: write back (endpoints t=0, t=255 never touched -> pinned) ----
#pragma unroll
        for (int k = 0; k < 8; ++k) {
            const int t = 1 + wv + 32 * k;
            if (t <= T_PTS - 2) p_lds[t * DIM + lane] = pnew[k];
        }
        __syncthreads();
    }

    // ---- drain LDS -> global output ----
    {
        float* g = out + base + (size_t)tid * 8;
        float* l = &p_lds[tid * 8];
#if defined(HAVE_ASYNC_ST)
        __builtin_amdgcn_global_store_async_from_lds_b128((g_v4i*)g, (l_v4i*)l, 0,  0);
        __builtin_amdgcn_global_store_async_from_lds_b128((g_v4i*)g, (l_v4i*)l, 16, 0);
        WAIT_ASYNC();
#else
        float4* g4 = (float4*)g;
        const float4* l4 = (const float4*)l;
        g4[0] = l4[0];
        g4[1] = l4[1];
#endif
    }
}

extern "C" void kernel_launch(void* const* d_in, const int* in_sizes, int n_in,
                              void* d_out, int out_size, void* d_ws, size_t ws_size,
                              hipStream_t stream) {
    (void)n_in; (void)d_ws; (void)ws_size; (void)out_size;
    const float* path   = (const float*)d_in[0];
    // d_in[1] = metric_field, d_in[2] = affordance: unused by the reference cost
    const float* step_p = (const float*)d_in[3];
    const float* mom_p  = (const float*)d_in[4];
    float* out = (float*)d_out;

    const int blocks = in_sizes[0] / (T_PTS * DIM);   // = batch B (32)
    GeodesicSolver_kernel<<<dim3(blocks), dim3(1024), 0, stream>>>(path, step_p, mom_p, out);
}